// GraphConvolution_layer_39462159515770
// MI455X (gfx1250) — compile-verified
//
#include <hip/hip_runtime.h>

typedef __attribute__((ext_vector_type(2))) float v2f;
typedef __attribute__((ext_vector_type(8))) float v8f;

#define N_NODES   100000
#define D_IN      512
#define D_OUT     128
#define N_EDGES   640000
#define LDS_PITCH 516   // 512 + 4 floats pad: row stride % 64 banks == 4 -> conflict-free

// ---------------------------------------------------------------------------
// Kernel 1: h = X @ W using V_WMMA_F32_16X16X4_F32 (full f32 precision).
// Block = 256 threads (8 waves). Block handles rows [r0, r0+16), each wave
// owns one 16-wide column tile (8 * 16 = 128 = D_OUT). K loop steps by 4.
// ---------------------------------------------------------------------------
__launch_bounds__(256)
__global__ void gemm_xw_wmma(const float* __restrict__ X,
                             const float* __restrict__ W,
                             float* __restrict__ H) {
  __shared__ float As[16 * LDS_PITCH];   // 33 KB of the 320 KB/WGP LDS
  const int r0 = blockIdx.x * 16;

  // Cooperative stage of the 16x512 A tile: 2048 float4 slots / 256 threads.
  {
    const int t = threadIdx.x;
#pragma unroll
    for (int i = 0; i < 8; ++i) {
      const int slot = t + 256 * i;      // 0..2047
      const int row  = slot >> 7;        // slot / 128
      const int c4   = slot & 127;       // float4 column
      const float4 v =
          *(const float4*)(X + (size_t)(r0 + row) * D_IN + (size_t)c4 * 4);
      *(float4*)(&As[row * LDS_PITCH + c4 * 4]) = v;   // 16B-aligned (2064*row)
    }
  }
  __syncthreads();

  const int lane  = threadIdx.x & 31;
  const int wave  = threadIdx.x >> 5;          // 0..7
  const int col   = wave * 16 + (lane & 15);   // output column (N)
  const int arow  = lane & 15;                 // A row (M) for this lane
  const int khalf = (lane >> 4) * 2;           // lanes 16..31 hold K+2,K+3

  v8f acc = {};
#pragma unroll 4
  for (int k = 0; k < D_IN; k += 4) {
    // A 16x4 f32 layout: lane(0-15)=M, VGPR0=K[k+khalf], VGPR1=K[k+khalf+1]
    const v2f a = *(const v2f*)(&As[arow * LDS_PITCH + k + khalf]);
    // B 4x16 f32 layout mirrors A: lane%16 = N, same K split across lane halves
    v2f b;
    b.x = W[(size_t)(k + khalf) * D_OUT + col];
    b.y = W[(size_t)(k + khalf + 1) * D_OUT + col];
    // 8 args: (neg_a, A, neg_b, B, c_mod, C, reuse_a, reuse_b)
    acc = __builtin_amdgcn_wmma_f32_16x16x4_f32(
        false, a, false, b, (short)0, acc, false, false);
  }

  // C/D layout: VGPR j -> M = j (lanes 0-15) or j+8 (lanes 16-31), N = lane%16
  const int rbase = r0 + (lane >> 4) * 8;
#pragma unroll
  for (int j = 0; j < 8; ++j) {
    H[(size_t)(rbase + j) * D_OUT + col] = acc[j];
  }
}

// ---------------------------------------------------------------------------
// Kernel 2: out[i, c] = bias[c]  (bias is added post-aggregation in the
// reference; seeding the accumulator with it is equivalent). Re-done every
// launch since the harness does not re-initialize d_out between replays.
// ---------------------------------------------------------------------------
__launch_bounds__(256)
__global__ void init_out_bias(float* __restrict__ out,
                              const float* __restrict__ bias) {
  const size_t i = (size_t)blockIdx.x * 256 + threadIdx.x;  // < N_NODES*D_OUT
  out[i] = bias[i & (D_OUT - 1)];
}

// ---------------------------------------------------------------------------
// Kernel 3: out[dst] += w_e * h[src]. One channel per thread, 2 edges per
// block. h and out both fit in the 192 MB L2, so the random gather + f32
// atomics stay on-die. unsafeAtomicAdd -> global_atomic_add_f32 (no CAS loop).
// ---------------------------------------------------------------------------
__launch_bounds__(256)
__global__ void edge_scatter(const long long* __restrict__ esrc,
                             const long long* __restrict__ edst,
                             const float* __restrict__ ew,
                             const float* __restrict__ H,
                             float* __restrict__ out) {
  const int e = blockIdx.x * 2 + (threadIdx.x >> 7);
  const int c = threadIdx.x & (D_OUT - 1);
  const long long s = esrc[e];
  const long long d = edst[e];
  const float     w = ew[e];
  const float     v = w * H[(size_t)s * D_OUT + c];
  unsafeAtomicAdd(out + (size_t)d * D_OUT + c, v);
}

// ---------------------------------------------------------------------------
extern "C" void kernel_launch(void* const* d_in, const int* in_sizes, int n_in,
                              void* d_out, int out_size, void* d_ws, size_t ws_size,
                              hipStream_t stream) {
  const float*     features = (const float*)d_in[0];      // [100000,512] f32
  const long long* esrc     = (const long long*)d_in[1];  // [640000] i64
  const long long* edst     = (const long long*)d_in[2];  // [640000] i64
  const float*     ew       = (const float*)d_in[3];      // [640000] f32
  const float*     W        = (const float*)d_in[4];      // [512,128] f32
  const float*     bias     = (const float*)d_in[5];      // [128] f32
  float* out = (float*)d_out;                             // [100000,128] f32
  float* H   = (float*)d_ws;                              // 51.2 MB scratch

  (void)in_sizes; (void)n_in; (void)out_size; (void)ws_size;

  gemm_xw_wmma<<<N_NODES / 16, 256, 0, stream>>>(features, W, H);
  init_out_bias<<<(N_NODES * D_OUT) / 256, 256, 0, stream>>>(out, bias);
  edge_scatter<<<N_EDGES / 2, 256, 0, stream>>>(esrc, edst, ew, H, out);
}